// MultiPartLoss_43181601194434
// MI455X (gfx1250) — compile-verified
//
#include <hip/hip_runtime.h>

typedef __attribute__((ext_vector_type(2))) float v2f;
typedef __attribute__((ext_vector_type(8))) float v8f;

#define NSS    49
#define NB     2
#define NC     20
#define NF     30
#define COORDW 5.0f
#define NOOBJW 0.5f
#define BLK    256

// ---------------------------------------------------------------------------
// Kernel 1: one thread per grid cell. Streams preds/targets (fully coalesced
// b64 loads), computes the per-cell loss, reduces per-wave (wave32 shfl) and
// per-block, writes one partial sum per block to d_ws.
// ---------------------------------------------------------------------------
__global__ __launch_bounds__(BLK) void yolo_loss_partial(
    const float* __restrict__ preds,
    const float* __restrict__ targets,
    float* __restrict__ partials,
    int ncells)
{
    const int cell = blockIdx.x * BLK + threadIdx.x;
    float loss = 0.0f;

    if (cell < ncells) {
        // cell base = cell * 120 bytes -> always 8-byte aligned: use float2 loads
        const float2* p2 = reinterpret_cast<const float2*>(preds   + (size_t)cell * NF);
        const float2* t2 = reinterpret_cast<const float2*>(targets + (size_t)cell * NF);
        float pf[NF], tf[NF];
#pragma unroll
        for (int i = 0; i < NF / 2; ++i) { float2 v = p2[i]; pf[2*i] = v.x; pf[2*i+1] = v.y; }
#pragma unroll
        for (int i = 0; i < NF / 2; ++i) { float2 v = t2[i]; tf[2*i] = v.x; tf[2*i+1] = v.y; }

        // 1) no-object confidence term (all cells, both boxes)
        float dc0 = pf[NC]     - tf[NC];
        float dc1 = pf[NC + 1] - tf[NC + 1];
        loss = NOOBJW * (dc0 * dc0 + dc1 * dc1);

        // obj = t_conf[...,0] > 0
        if (tf[NC] > 0.0f) {
            // 2) class-probability term
            float s = 0.0f;
#pragma unroll
            for (int i = 0; i < NC; ++i) { float d = pf[i] - tf[i]; s += d * d; }
            loss += s;

            // 3) IoU of each predicted box vs its target box
            float iou[NB];
#pragma unroll
            for (int k = 0; k < NB; ++k) {
                const int o = NC + NB + 4 * k;
                float px = pf[o], py = pf[o+1], pw = pf[o+2], ph = pf[o+3];
                float tx = tf[o], ty = tf[o+1], tw = tf[o+2], th = tf[o+3];
                float xA = fmaxf(px - 0.5f * pw, tx - 0.5f * tw);
                float yA = fmaxf(py - 0.5f * ph, ty - 0.5f * th);
                float xB = fminf(px + 0.5f * pw, tx + 0.5f * tw);
                float yB = fminf(py + 0.5f * ph, ty + 0.5f * th);
                float inter = fmaxf(0.0f, xB - xA) * fmaxf(0.0f, yB - yA);
                iou[k] = inter / (pw * ph + tw * th - inter);
            }
            // jnp.argmax picks the FIRST max -> best=1 only on strict >
            const int best = (iou[1] > iou[0]) ? 1 : 0;

            // 4) responsible-box confidence term
            float pc = pf[NC + best], tc = tf[NC + best];
            loss += (1.0f - NOOBJW) * (pc - tc) * (pc - tc);

            // 5) coordinate term
            const int o = NC + NB + 4 * best;
            float dx = pf[o]     - tf[o];
            float dy = pf[o + 1] - tf[o + 1];
            float dw = sqrtf(pf[o + 2]) - sqrtf(tf[o + 2]);
            float dh = sqrtf(pf[o + 3]) - sqrtf(tf[o + 3]);
            loss += COORDW * (dx * dx + dy * dy + dw * dw + dh * dh);
        }
    }

    // wave32 reduction
#pragma unroll
    for (int off = 16; off > 0; off >>= 1)
        loss += __shfl_xor(loss, off, 32);

    __shared__ float wsum[BLK / 32];
    const int lane = threadIdx.x & 31;
    const int wave = threadIdx.x >> 5;
    if (lane == 0) wsum[wave] = loss;
    __syncthreads();
    if (threadIdx.x == 0) {
        float b = 0.0f;
#pragma unroll
        for (int w = 0; w < BLK / 32; ++w) b += wsum[w];
        partials[blockIdx.x] = b;
    }
}

// ---------------------------------------------------------------------------
// Kernel 2: single block reduces all block partials. The last 32-wide stage
// uses v_wmma_f32_16x16x4_f32: A = lane values in the (M, K=0|2) slots of the
// 16x4 f32 A layout, B = ones(4x16) -> D rows hold exact A-row sums
// (v[M] + v[M+16]); summing the 8 accumulator VGPRs per lane plus one
// shfl_xor(16) gives the exact fp32 total.
// ---------------------------------------------------------------------------
__global__ __launch_bounds__(BLK) void yolo_loss_final(
    const float* __restrict__ partials,
    int n,
    float inv_n_img,
    float* __restrict__ out)
{
    __shared__ float sdata[BLK];
    float s = 0.0f;
    for (int i = threadIdx.x; i < n; i += BLK) s += partials[i];
    sdata[threadIdx.x] = s;
    __syncthreads();

    if (threadIdx.x < 32) {   // whole wave 0 active -> EXEC all ones for WMMA
        float v = 0.0f;
#pragma unroll
        for (int j = 0; j < BLK / 32; ++j) v += sdata[threadIdx.x + 32 * j];

        // A (16x4 f32): VGPR0 = {lanes 0-15: K=0, lanes 16-31: K=2}, VGPR1 = 0
        v2f a; a.x = v;    a.y = 0.0f;
        v2f b; b.x = 1.0f; b.y = 1.0f;   // B (4x16) = all ones
        v8f c = {};
        c = __builtin_amdgcn_wmma_f32_16x16x4_f32(
                /*neg_a=*/false, a, /*neg_b=*/false, b,
                /*c_mod=*/(short)0, c, /*reuse_a=*/false, /*reuse_b=*/false);

        // per lane: sum of 8 row-sums (rows 0..7 for lanes<16, rows 8..15 else)
        float h = c[0] + c[1] + c[2] + c[3] + c[4] + c[5] + c[6] + c[7];
        float tot = h + __shfl_xor(h, 16, 32);
        if (threadIdx.x == 0) out[0] = tot * inv_n_img;
    }
}

// ---------------------------------------------------------------------------
extern "C" void kernel_launch(void* const* d_in, const int* in_sizes, int n_in,
                              void* d_out, int out_size, void* d_ws, size_t ws_size,
                              hipStream_t stream)
{
    const float* preds   = (const float*)d_in[0];
    const float* targets = (const float*)d_in[1];

    const int ncells = in_sizes[0] / NF;           // N_IMG * 49
    const int n_img  = ncells / NSS;
    const int blocks = (ncells + BLK - 1) / BLK;

    float* partials = (float*)d_ws;

    yolo_loss_partial<<<blocks, BLK, 0, stream>>>(preds, targets, partials, ncells);
    yolo_loss_final<<<1, BLK, 0, stream>>>(partials, blocks, 1.0f / (float)n_img,
                                           (float*)d_out);
}